// RefiedKBQA_87445534146881
// MI455X (gfx1250) — compile-verified
//
#include <hip/hip_runtime.h>
#include <hip/hip_bf16.h>

// ---------------------------------------------------------------------------
// RefiedKBQA 3-hop sparse KB propagation for MI455X (gfx1250, wave32).
//
//  x:(64,200000) f32   q:(64,300)   W*: (300,200)  b*: (200,)
//  subj/rel/obj: (1e6,) int32       n_hop = 3
//
//  hop:  r = q@W + b           (WMMA f32 16x16x4, the only dense matmul)
//        m = x[:,subj]*r[:,rel]; x' = scatter_add(m, obj)   (L2-resident)
// ---------------------------------------------------------------------------

#define KBQA_NE 200000
#define KBQA_B  64
#define KBQA_NR 200
#define KBQA_K  300

typedef __attribute__((ext_vector_type(2))) float v2f;
typedef __attribute__((ext_vector_type(8))) float v8f;

// ---------------- transpose x (B, N_E) -> xT (N_E, B), LDS tiled -----------
__global__ void kbqa_transpose_in(const float* __restrict__ x,
                                  float* __restrict__ xT, int n_e) {
    __shared__ float tile[64][65];
    int e0 = blockIdx.x * 64;
    for (int i = threadIdx.x; i < 64 * 64; i += blockDim.x) {
        int b = i >> 6, ec = i & 63;          // tid fast over ec -> coalesced read
        int e = e0 + ec;
        tile[ec][b] = (e < n_e) ? x[(size_t)b * n_e + e] : 0.0f;
    }
    __syncthreads();
    for (int i = threadIdx.x; i < 64 * 64; i += blockDim.x) {
        int ec = i >> 6, b = i & 63;          // tid fast over b -> coalesced write
        int e = e0 + ec;
        if (e < n_e) xT[(size_t)e * 64 + b] = tile[ec][b];
    }
}

// ---------------- transpose xT (N_E, B) -> out (B, N_E) --------------------
__global__ void kbqa_transpose_out(const float* __restrict__ xT,
                                   float* __restrict__ out, int n_e) {
    __shared__ float tile[64][65];
    int e0 = blockIdx.x * 64;
    for (int i = threadIdx.x; i < 64 * 64; i += blockDim.x) {
        int ec = i >> 6, b = i & 63;          // tid fast over b -> coalesced read
        int e = e0 + ec;
        tile[b][ec] = (e < n_e) ? xT[(size_t)e * 64 + b] : 0.0f;
    }
    __syncthreads();
    for (int i = threadIdx.x; i < 64 * 64; i += blockDim.x) {
        int b = i >> 6, ec = i & 63;          // tid fast over ec -> coalesced write
        int e = e0 + ec;
        if (e < n_e) out[(size_t)b * n_e + e] = tile[b][ec];
    }
}

// ---------------- r = q @ W + bias via V_WMMA_F32_16X16X4_F32 --------------
// One wave32 per 16x16 output tile. M=64 (batch), N=200 (relations), K=300.
// Output stored transposed: rT[n*64 + m] so rel-gather columns are contiguous.
//
// f32 WMMA layouts (ISA 7.12.2):
//   A 16x4 : lanes 0-15 rows, VGPR{0,1} = K{0,1}; lanes 16-31 = K{2,3}
//   B 4x16 : lanes 0-15 cols, VGPR{0,1} = K{0,1}; lanes 16-31 = K{2,3}
//   C 16x16: VGPR i -> row i (lanes 0-15) / row i+8 (lanes 16-31), col = lane&15
__global__ void kbqa_gemm_r_wmma(const float* __restrict__ q,   // (64,300)
                                 const float* __restrict__ W,   // (300,200)
                                 const float* __restrict__ bias,// (200,)
                                 float* __restrict__ rT) {      // (200,64)
    int lane  = threadIdx.x;          // blockDim == 32, full wave, EXEC all 1s
    int mt    = blockIdx.x & 3;       // 4 M-tiles (64 rows)
    int nt    = blockIdx.x >> 2;      // 13 N-tiles (208 cols, last partial)
    int half  = lane >> 4;            // 0 | 1
    int l15   = lane & 15;
    int row   = mt * 16 + l15;        // A row (valid: < 64)
    int colR  = nt * 16 + l15;        // B/D col (may be >= 200)
    int col   = colR < KBQA_NR ? colR : (KBQA_NR - 1);  // clamp loads only

    v8f c = {};
    for (int kk = 0; kk < KBQA_K; kk += 4) {
        int kb = kk + half * 2;
        v2f a = { q[row * KBQA_K + kb], q[row * KBQA_K + kb + 1] };
        v2f b = { W[kb * KBQA_NR + col], W[(kb + 1) * KBQA_NR + col] };
        c = __builtin_amdgcn_wmma_f32_16x16x4_f32(
                /*neg_a=*/false, a, /*neg_b=*/false, b,
                /*c_mod=*/(short)0, c, /*reuse_a=*/false, /*reuse_b=*/false);
    }

    if (colR < KBQA_NR) {
        float bv = bias[colR];
#pragma unroll
        for (int i = 0; i < 8; ++i) {
            int m = mt * 16 + i + half * 8;
            rT[colR * KBQA_B + m] = c[i] + bv;
        }
    }
}

// ---------------- heavy step: gather * rel, scatter-add over obj -----------
// One wave per triple per iteration; lane handles 2 batch elements (float2).
// rT staged in LDS (51.2 KB of the 320 KB WGP pool).
__global__ void kbqa_scatter_step(const float* __restrict__ xT_src,
                                  float* __restrict__ xT_dst,
                                  const float* __restrict__ rT,
                                  const int* __restrict__ subj,
                                  const int* __restrict__ rel,
                                  const int* __restrict__ obj,
                                  int n_t) {
    __shared__ float lds_r[KBQA_NR * KBQA_B];   // 12800 f32 = 51.2 KB
    for (int i = threadIdx.x; i < KBQA_NR * KBQA_B; i += blockDim.x)
        lds_r[i] = rT[i];
    __syncthreads();

    const int lane   = threadIdx.x & 31;
    const int wib    = threadIdx.x >> 5;
    const int wpb    = blockDim.x >> 5;
    const int gwave  = blockIdx.x * wpb + wib;
    const int nwaves = gridDim.x * wpb;
    const int bi     = lane * 2;                // batch pair per lane

    for (int t = gwave; t < n_t; t += nwaves) {
        // per-wave-uniform triple indices -> force into SGPRs
        int s  = __builtin_amdgcn_readfirstlane(subj[t]);
        int rl = __builtin_amdgcn_readfirstlane(rel[t]);
        int o  = __builtin_amdgcn_readfirstlane(obj[t]);

        int tp = t + nwaves * 8;                // stay ahead on index streams
        if (tp < n_t) {
            __builtin_prefetch(&subj[tp], 0, 0);
            __builtin_prefetch(&rel[tp],  0, 0);
            __builtin_prefetch(&obj[tp],  0, 0);
        }

        float2 xv = *(const float2*)(xT_src + (size_t)s * KBQA_B + bi);
        float2 rv = *(const float2*)(&lds_r[rl * KBQA_B + bi]);

        float* dst = xT_dst + (size_t)o * KBQA_B + bi;
        atomicAdd(dst + 0, xv.x * rv.x);        // global_atomic_add_f32, L2-resident
        atomicAdd(dst + 1, xv.y * rv.y);
    }
}

// ---------------------------------------------------------------------------
extern "C" void kernel_launch(void* const* d_in, const int* in_sizes, int n_in,
                              void* d_out, int out_size, void* d_ws, size_t ws_size,
                              hipStream_t stream) {
    const float* x    = (const float*)d_in[0];
    const float* q    = (const float*)d_in[1];
    const float* Ws[3] = { (const float*)d_in[2], (const float*)d_in[4], (const float*)d_in[6] };
    const float* bs[3] = { (const float*)d_in[3], (const float*)d_in[5], (const float*)d_in[7] };
    const int* subj = (const int*)d_in[8];
    const int* rel  = (const int*)d_in[9];
    const int* obj  = (const int*)d_in[10];
    const int n_t   = in_sizes[8];

    const size_t xT_elems = (size_t)KBQA_NE * KBQA_B;   // 12.8M f32 = 51.2 MB
    float* xT_a = (float*)d_ws;
    float* xT_b = xT_a + xT_elems;
    float* rT   = xT_b + xT_elems;                       // +51.2 KB

    const int tr_blocks = (KBQA_NE + 63) / 64;           // 3125

    kbqa_transpose_in<<<tr_blocks, 256, 0, stream>>>(x, xT_a, KBQA_NE);

    float* cur = xT_a;
    float* nxt = xT_b;
    for (int h = 0; h < 3; ++h) {                        // n_hop == 3
        kbqa_gemm_r_wmma<<<4 * 13, 32, 0, stream>>>(q, Ws[h], bs[h], rT);
        hipMemsetAsync(nxt, 0, xT_elems * sizeof(float), stream);
        kbqa_scatter_step<<<1024, 256, 0, stream>>>(cur, nxt, rT, subj, rel, obj, n_t);
        float* tmp = cur; cur = nxt; nxt = tmp;
    }

    kbqa_transpose_out<<<tr_blocks, 256, 0, stream>>>(cur, (float*)d_out, KBQA_NE);
}